// GraphAutoencoder_3788161155350
// MI455X (gfx1250) — compile-verified
//
#include <hip/hip_runtime.h>
#include <math.h>

// ---------------- problem constants (from reference) ----------------
#define NN      50000      // nodes (divisible by 16: 3125 row tiles)
#define E_IN    800000     // input edges
#define E_TOT   850000     // + self loops
#define NEGSL   0.2f       // leaky relu slope for attention logits

typedef __attribute__((ext_vector_type(2))) float v2f;
typedef __attribute__((ext_vector_type(8))) float v8f;

// ---------------- WMMA fp32 GEMM: H[N x don] = A[N x di] @ W[di x don] ----
// One wave computes one 16x16 output tile using V_WMMA_F32_16X16X4_F32,
// K-loop in steps of 4.  A layout: lane(m=lane&15) holds K = k + 2*(lane>>4)
// and +1; B layout mirrors it; C/D: vgpr v -> row v + 8*(lane>>4), col lane&15.
__global__ __launch_bounds__(128) void gat_gemm_wmma(
    const float* __restrict__ A, const float* __restrict__ W,
    float* __restrict__ H, int di, int don) {
  const int lane   = threadIdx.x & 31;
  const int wave   = threadIdx.x >> 5;
  const int tilesN = don >> 4;
  const int total  = (NN >> 4) * tilesN;
  const int tile   = blockIdx.x * 4 + wave;
  if (tile >= total) return;                 // wave-uniform: EXEC stays all-1
  const int tm = tile / tilesN;
  const int tn = tile - tm * tilesN;
  const int m  = lane & 15;                  // row-in-tile for A, col for B/C
  const int kb = (lane >> 4) << 1;           // K sub-offset per half-wave
  const int col = (tn << 4) + m;

  const float* arow = A + (size_t)((tm << 4) + m) * di;
  v8f c = {};
  for (int k = 0; k < di; k += 4) {
    v2f a, b;
    a.x = arow[k + kb];
    a.y = arow[k + kb + 1];
    b.x = W[(size_t)(k + kb)     * don + col];
    b.y = W[(size_t)(k + kb + 1) * don + col];
    c = __builtin_amdgcn_wmma_f32_16x16x4_f32(
        /*neg_a=*/false, a, /*neg_b=*/false, b,
        /*c_mod=*/(short)0, c, /*reuse_a=*/false, /*reuse_b=*/false);
  }
  const int rbase = (tm << 4) + ((lane >> 4) << 3);
  for (int v = 0; v < 8; ++v)
    H[(size_t)(rbase + v) * don + col] = c[v];
}

// ------------- per-node: s = h.a_src, d = h.a_dst; init m/denom/acc -------
__global__ void gat_node_prep(
    const float* __restrict__ H, const float* __restrict__ asrc,
    const float* __restrict__ adst, const float* __restrict__ bias,
    float* __restrict__ s, float* __restrict__ d,
    float* __restrict__ m, float* __restrict__ denom,
    float* __restrict__ acc, int don) {
  int i = blockIdx.x * blockDim.x + threadIdx.x;
  if (i >= NN) return;
  const float* h = H + (size_t)i * don;
  float ss = 0.f, dd = 0.f;
  for (int c = 0; c < don; ++c) { float hv = h[c]; ss += hv * asrc[c]; dd += hv * adst[c]; }
  s[i] = ss; d[i] = dd;
  m[i] = -INFINITY; denom[i] = 0.f;
  float* a = acc + (size_t)i * don;
  for (int c = 0; c < don; ++c) a[c] = bias[c];   // out = segsum + b
}

// ordered-int trick: correct atomic float max (values may be negative)
__device__ __forceinline__ void atomicMaxF(float* addr, float v) {
  if (v >= 0.f) atomicMax((int*)addr, __float_as_int(v));
  else          atomicMin((unsigned int*)addr, (unsigned int)__float_as_int(v));
}

__device__ __forceinline__ void edge_ends(int e, const int* __restrict__ esrc,
                                          const int* __restrict__ edst,
                                          int& si, int& di_) {
  if (e < E_IN) { si = esrc[e]; di_ = edst[e]; }
  else          { si = e - E_IN; di_ = si; }      // self loops appended
}

// ------------- pass 1: logits (leaky-relu) + segment max ------------------
__global__ void gat_edge_max(
    const int* __restrict__ esrc, const int* __restrict__ edst,
    const float* __restrict__ s, const float* __restrict__ d,
    float* __restrict__ m, float* __restrict__ ebuf) {
  int e = blockIdx.x * blockDim.x + threadIdx.x;
  if (e >= E_TOT) return;
  int si, di_; edge_ends(e, esrc, edst, si, di_);
  float v = s[si] + d[di_];
  v = v > 0.f ? v : NEGSL * v;
  ebuf[e] = v;
  atomicMaxF(&m[di_], v);
}

// ------------- pass 2: exp(e - m[dst]) + segment sum ----------------------
__global__ void gat_edge_expsum(
    const int* __restrict__ esrc, const int* __restrict__ edst,
    const float* __restrict__ m, float* __restrict__ ebuf,
    float* __restrict__ denom) {
  int e = blockIdx.x * blockDim.x + threadIdx.x;
  if (e >= E_TOT) return;
  int si, di_; edge_ends(e, esrc, edst, si, di_);
  float ex = __expf(ebuf[e] - m[di_]);
  ebuf[e] = ex;
  atomicAdd(&denom[di_], ex);
}

// ------------- pass 3: acc[dst] += alpha * h[src]  (thread = edge x 4ch) --
__global__ void gat_edge_aggregate(
    const int* __restrict__ esrc, const int* __restrict__ edst,
    const float* __restrict__ H, const float* __restrict__ ebuf,
    const float* __restrict__ denom, float* __restrict__ acc, int don) {
  int t  = blockIdx.x * blockDim.x + threadIdx.x;
  int cg = don >> 2;                         // float4 channel groups
  int e  = t / cg;
  if (e >= E_TOT) return;
  int c4 = (t - e * cg) << 2;
  int si, di_; edge_ends(e, esrc, edst, si, di_);
  float alpha = ebuf[e] / (denom[di_] + 1e-16f);
  const float4 hv = *(const float4*)(H + (size_t)si * don + c4);
  float* a = acc + (size_t)di_ * don + c4;
  atomicAdd(a + 0, alpha * hv.x);
  atomicAdd(a + 1, alpha * hv.y);
  atomicAdd(a + 2, alpha * hv.z);
  atomicAdd(a + 3, alpha * hv.w);
}

// ------------- epilogue: relu into next-layer input (+ optional raw copy) -
__global__ void gat_relu_copy(const float* __restrict__ acc,
                              float* __restrict__ hin,
                              float* __restrict__ raw, int count) {
  int i = blockIdx.x * blockDim.x + threadIdx.x;
  if (i >= count) return;
  float v = acc[i];
  if (raw) raw[i] = v;                       // layer-2 pre-relu output
  hin[i] = v > 0.f ? v : 0.f;
}

// --------------------------------------------------------------------------
extern "C" void kernel_launch(void* const* d_in, const int* in_sizes, int n_in,
                              void* d_out, int out_size, void* d_ws, size_t ws_size,
                              hipStream_t stream) {
  (void)in_sizes; (void)n_in; (void)out_size; (void)ws_size;

  const float* x    = (const float*)d_in[0];
  const int*   ei   = (const int*)d_in[1];       // [2, 800000] row-major
  const int*   esrc = ei;
  const int*   edst = ei + E_IN;

  const float *w[4], *as_[4], *ad_[4], *bs[4];
  for (int l = 0; l < 4; ++l) {
    w[l]   = (const float*)d_in[2 + 4 * l];
    as_[l] = (const float*)d_in[3 + 4 * l];
    ad_[l] = (const float*)d_in[4 + 4 * l];
    bs[l]  = (const float*)d_in[5 + 4 * l];
  }
  const int din[4]  = {64, 64, 32, 64};
  const int dout[4] = {64, 32, 64, 64};

  // workspace layout (floats); total ~10.7M floats (~43 MB)
  float* ws    = (float*)d_ws;
  float* h     = ws;                      // N*64
  float* acc   = h   + (size_t)NN * 64;   // N*64
  float* hin   = acc + (size_t)NN * 64;   // N*64
  float* sv    = hin + (size_t)NN * 64;   // N
  float* dv    = sv + NN;                 // N
  float* mv    = dv + NN;                 // N
  float* denom = mv + NN;                 // N
  float* ebuf  = denom + NN;              // E_TOT

  float* out_final = (float*)d_out;             // [N,64]
  float* out_h     = out_final + (size_t)NN * 64; // [N,32] layer-2 pre-relu

  const float* cur = x;
  for (int l = 0; l < 4; ++l) {
    float* accbuf = (l == 3) ? out_final : acc;

    // dense projection via WMMA fp32
    int tiles = (NN >> 4) * (dout[l] >> 4);
    gat_gemm_wmma<<<(tiles + 3) / 4, 128, 0, stream>>>(cur, w[l], h, din[l], dout[l]);

    // per-node logits + init segment buffers + init acc with bias
    gat_node_prep<<<(NN + 255) / 256, 256, 0, stream>>>(
        h, as_[l], ad_[l], bs[l], sv, dv, mv, denom, accbuf, dout[l]);

    // segment softmax (3 passes) + weighted scatter-add
    gat_edge_max<<<(E_TOT + 255) / 256, 256, 0, stream>>>(esrc, edst, sv, dv, mv, ebuf);
    gat_edge_expsum<<<(E_TOT + 255) / 256, 256, 0, stream>>>(esrc, edst, mv, ebuf, denom);
    long long aggT = (long long)E_TOT * (dout[l] >> 2);
    gat_edge_aggregate<<<(unsigned)((aggT + 255) / 256), 256, 0, stream>>>(
        esrc, edst, h, ebuf, denom, accbuf, dout[l]);

    // activation / output copies
    if (l == 0 || l == 2) {
      gat_relu_copy<<<((NN * dout[l]) + 255) / 256, 256, 0, stream>>>(
          acc, hin, nullptr, NN * dout[l]);
      cur = hin;
    } else if (l == 1) {
      gat_relu_copy<<<((NN * 32) + 255) / 256, 256, 0, stream>>>(
          acc, hin, out_h, NN * 32);
      cur = hin;
    }
    // l == 3: final layer wrote directly into d_out, no activation
  }
}